// GCN_79517024518684
// MI455X (gfx1250) — compile-verified
//
#include <hip/hip_runtime.h>

typedef __attribute__((ext_vector_type(2))) float v2f;
typedef __attribute__((ext_vector_type(8))) float v8f;

#define INF_  7
#define KPAD  8
#define HDIM  32
#define CDIM  2
#define EPSV  1e-5f

// ---------------------------------------------------------------------------
// Degree / normalization
// ---------------------------------------------------------------------------
__global__ void deg_init_kernel(float* __restrict__ deg, long long n) {
    long long i = (long long)blockIdx.x * blockDim.x + threadIdx.x;
    if (i < n) deg[i] = 1.0f;                 // self-loop weight
}

__global__ void deg_edge_kernel(const int* __restrict__ col, const float* __restrict__ ea,
                                float* __restrict__ deg, long long ne) {
    long long e = (long long)blockIdx.x * blockDim.x + threadIdx.x;
    if (e < ne) atomicAdd(&deg[col[e]], ea[e]);
}

__global__ void deg_to_dinv_kernel(float* __restrict__ deg, long long n) {
    long long i = (long long)blockIdx.x * blockDim.x + threadIdx.x;
    if (i < n) {
        float d = deg[i];
        deg[i] = (d > 0.0f) ? rsqrtf(fmaxf(d, EPSV)) : 0.0f;
    }
}

// Zero-pad [N,7] node features to [N,8] so the GEMM K-loop is branch-free.
__global__ void pad_x_kernel(const float* __restrict__ x, float* __restrict__ xp,
                             long long n) {
    long long t = (long long)blockIdx.x * blockDim.x + threadIdx.x;
    if (t >= n * KPAD) return;
    long long i = t >> 3;
    int k = (int)(t & 7);
    xp[t] = (k < INF_) ? x[i * INF_ + k] : 0.0f;
}

// ---------------------------------------------------------------------------
// Dense transform  Y[rows x 32] = X[rows x K] @ W[KSRC x 32] (+bias), fp32 WMMA.
// K is compile-time and a multiple of 4: no k-guards, A loads are aligned
// global_load_b64, B comes from LDS. Row tail handled by clamp (A row m only
// feeds D row m, and OOB D rows are never stored) + a full-tile store fast path.
// ---------------------------------------------------------------------------
template <int K, int KSRC>
__global__ __launch_bounds__(256) void gemm_wmma_kernel(
    const float* __restrict__ X, const float* __restrict__ W,
    const float* __restrict__ bias, float* __restrict__ Y, int nrows) {
    static_assert((K & 3) == 0, "K must be a multiple of 4");
    __shared__ float sW[K * HDIM];
    for (int idx = threadIdx.x; idx < K * HDIM; idx += 256)
        sW[idx] = (idx < KSRC * HDIM) ? W[idx] : 0.0f;
    __syncthreads();

    const int lane = threadIdx.x & 31;
    const int wave = threadIdx.x >> 5;
    const int tile = blockIdx.x * 8 + wave;
    const int half = lane >> 4;
    const int n    = lane & 15;                 // B/D column within 16-wide half
    const int arow = tile * 16 + (lane & 15);   // A row
    const int acl  = min(arow, nrows - 1);      // exact: see note above
    const float* __restrict__ Xr = X + (long long)acl * K;

    v8f c0 = {};  // columns 0..15
    v8f c1 = {};  // columns 16..31

    #pragma unroll
    for (int k0 = 0; k0 < K; k0 += 4) {
        const int kk = k0 + 2 * half;           // ISA A/B layout: v0<->kk, v1<->kk+1
        v2f a = *(const v2f*)(Xr + kk);         // 8B aligned (K%4==0, kk even)
        v2f b0, b1;
        b0.x = sW[kk       * HDIM + n];
        b0.y = sW[(kk + 1) * HDIM + n];
        b1.x = sW[kk       * HDIM + 16 + n];
        b1.y = sW[(kk + 1) * HDIM + 16 + n];
        c0 = __builtin_amdgcn_wmma_f32_16x16x4_f32(false, a, false, b0, (short)0, c0, false, false);
        c1 = __builtin_amdgcn_wmma_f32_16x16x4_f32(false, a, false, b1, (short)0, c1, false, false);
    }

    const float add0 = bias ? bias[n]      : 0.0f;
    const float add1 = bias ? bias[16 + n] : 0.0f;
    const long long base = (long long)tile * 16 + 8 * half;
    if ((tile + 1) * 16 <= nrows) {             // full tile: unguarded stores
        #pragma unroll
        for (int j = 0; j < 8; ++j) {
            Y[(base + j) * HDIM + n]      = c0[j] + add0;
            Y[(base + j) * HDIM + 16 + n] = c1[j] + add1;
        }
    } else {                                    // tail tile
        #pragma unroll
        for (int j = 0; j < 8; ++j) {
            if (base + j < nrows) {
                Y[(base + j) * HDIM + n]      = c0[j] + add0;
                Y[(base + j) * HDIM + 16 + n] = c1[j] + add1;
            }
        }
    }
}

// ---------------------------------------------------------------------------
// Conv: warp per node  out[i] = b + dinv[i]^2 * hw[i]   (self-loop + bias)
//       warp per edge  out[col] += dinv[row]*w*dinv[col] * hw[row]
// Edge id / indices forced uniform -> scalar loads, lanes carry features only.
// ---------------------------------------------------------------------------
__global__ __launch_bounds__(256) void conv_init_kernel(
    const float* __restrict__ hw, const float* __restrict__ dinv,
    const float* __restrict__ bias, float* __restrict__ out, int nrows) {
    int i = __builtin_amdgcn_readfirstlane(blockIdx.x * 8 + (threadIdx.x >> 5));
    if (i >= nrows) return;
    int f = threadIdx.x & 31;
    float d = dinv[i];
    out[(long long)i * HDIM + f] = bias[f] + d * d * hw[(long long)i * HDIM + f];
}

__global__ __launch_bounds__(256) void conv_edge_kernel(
    const int* __restrict__ row, const int* __restrict__ col,
    const float* __restrict__ ea, const float* __restrict__ dinv,
    const float* __restrict__ hw, float* __restrict__ out, int nedges) {
    int e = __builtin_amdgcn_readfirstlane(blockIdx.x * 8 + (threadIdx.x >> 5));
    if (e >= nedges) return;
    int f = threadIdx.x & 31;
    int r = __builtin_amdgcn_readfirstlane(row[e]);
    int c = __builtin_amdgcn_readfirstlane(col[e]);
    float nrm = dinv[r] * ea[e] * dinv[c];
    atomicAdd(&out[(long long)c * HDIM + f], nrm * hw[(long long)r * HDIM + f]);
}

// ---------------------------------------------------------------------------
// BatchNorm (training mode): per-feature sum/sumsq, then normalize + ReLU
// ---------------------------------------------------------------------------
__global__ __launch_bounds__(256) void stats_kernel(const float* __restrict__ h,
                                                    float* __restrict__ stats,
                                                    long long nrows) {
    __shared__ float ssum[8][32];
    __shared__ float ssq[8][32];
    const int f = threadIdx.x & 31;
    const int w = threadIdx.x >> 5;
    const long long gid   = (long long)blockIdx.x * blockDim.x + threadIdx.x;
    const long long warps = ((long long)gridDim.x * blockDim.x) >> 5;
    float s = 0.0f, s2 = 0.0f;
    for (long long i = gid >> 5; i < nrows; i += warps) {
        float v = h[i * HDIM + f];
        s += v; s2 += v * v;
    }
    ssum[w][f] = s; ssq[w][f] = s2;
    __syncthreads();
    if (w == 0) {
        float ts = 0.0f, t2 = 0.0f;
        #pragma unroll
        for (int j = 0; j < 8; ++j) { ts += ssum[j][f]; t2 += ssq[j][f]; }
        atomicAdd(&stats[f], ts);
        atomicAdd(&stats[HDIM + f], t2);
    }
}

__global__ void bn_relu_kernel(const float* __restrict__ x, float* __restrict__ y,
                               const float* __restrict__ stats,
                               const float* __restrict__ gamma, const float* __restrict__ beta,
                               long long nrows) {
    long long t = (long long)blockIdx.x * blockDim.x + threadIdx.x;
    if (t >= nrows * HDIM) return;
    int f = (int)(t & 31);
    float inv_n = 1.0f / (float)nrows;
    float mean  = stats[f] * inv_n;
    float var   = stats[HDIM + f] * inv_n - mean * mean;
    float v = (x[t] - mean) * rsqrtf(var + EPSV) * gamma[f] + beta[f];
    y[t] = fmaxf(v, 0.0f);
}

// ---------------------------------------------------------------------------
// Global add-pool (warp per node, scalar batch id) and classifier head
// ---------------------------------------------------------------------------
__global__ __launch_bounds__(256) void pool_kernel(
    const float* __restrict__ h, const int* __restrict__ batch,
    float* __restrict__ g, int nrows) {
    int i = __builtin_amdgcn_readfirstlane(blockIdx.x * 8 + (threadIdx.x >> 5));
    if (i >= nrows) return;
    int f = threadIdx.x & 31;
    int b = __builtin_amdgcn_readfirstlane(batch[i]);
    atomicAdd(&g[(long long)b * HDIM + f], h[(long long)i * HDIM + f]);
}

__global__ void head_kernel(const float* __restrict__ g2, const float* __restrict__ Wm1,
                            const float* __restrict__ bm1, float* __restrict__ out, int ng) {
    int t = blockIdx.x * blockDim.x + threadIdx.x;
    if (t >= ng * CDIM) return;
    int gi = t / CDIM, c = t % CDIM;
    float s = bm1[c];
    #pragma unroll
    for (int f = 0; f < HDIM; ++f) s += g2[gi * HDIM + f] * Wm1[f * CDIM + c];
    out[t] = s;
}

// ---------------------------------------------------------------------------
// Launch
// ---------------------------------------------------------------------------
static inline unsigned long long cdiv(unsigned long long a, unsigned long long b) {
    return (a + b - 1) / b;
}

extern "C" void kernel_launch(void* const* d_in, const int* in_sizes, int n_in,
                              void* d_out, int out_size, void* d_ws, size_t ws_size,
                              hipStream_t stream) {
    const long long N = in_sizes[0] / INF_;
    const long long E = in_sizes[1] / 2;
    const int       G = out_size / CDIM;

    const float* x     = (const float*)d_in[0];
    const int*   erow  = (const int*)d_in[1];
    const int*   ecol  = erow + E;
    const float* ea    = (const float*)d_in[2];
    const int*   batch = (const int*)d_in[3];
    const float* Wl[4] = {(const float*)d_in[4], (const float*)d_in[6],
                          (const float*)d_in[8], (const float*)d_in[10]};
    const float* bl[4] = {(const float*)d_in[5], (const float*)d_in[7],
                          (const float*)d_in[9], (const float*)d_in[11]};
    const float* bn_g  = (const float*)d_in[12];   // [3,32]
    const float* bn_b  = (const float*)d_in[13];
    const float* Wm0   = (const float*)d_in[14];
    const float* bm0   = (const float*)d_in[15];
    const float* bnm_g = (const float*)d_in[16];
    const float* bnm_b = (const float*)d_in[17];
    const float* Wm1   = (const float*)d_in[18];
    const float* bm1   = (const float*)d_in[19];

    // Workspace layout
    float* B0    = (float*)d_ws;                    // [N,32] conv out / activations
    float* B1    = B0 + (size_t)N * HDIM;           // [N,32] h@W staging
    float* dinv  = B1 + (size_t)N * HDIM;           // [N]
    float* stats = dinv + N;                        // [64] sum / sumsq
    float* gpool = stats + 2 * HDIM;                // [G,32]
    float* gw    = gpool + (size_t)G * HDIM;        // [G,32]
    float* xp    = gw + (size_t)G * HDIM;           // [N,8] padded layer-0 input

    const int T = 256;

    // ---- degree / dinv, padded input ----
    deg_init_kernel<<<(unsigned)cdiv(N, T), T, 0, stream>>>(dinv, N);
    deg_edge_kernel<<<(unsigned)cdiv(E, T), T, 0, stream>>>(ecol, ea, dinv, E);
    deg_to_dinv_kernel<<<(unsigned)cdiv(N, T), T, 0, stream>>>(dinv, N);
    pad_x_kernel<<<(unsigned)cdiv(N * KPAD, T), T, 0, stream>>>(x, xp, N);

    // ---- 4 GCN conv layers ----
    const unsigned gemmBlocksN = (unsigned)cdiv(cdiv(N, 16), 8);   // 8 waves/block
    const unsigned nodeBlocks  = (unsigned)cdiv(N, 8);             // warp per node
    const unsigned edgeBlocks  = (unsigned)cdiv(E, 8);             // warp per edge
    for (int layer = 0; layer < 4; ++layer) {
        if (layer == 0)
            gemm_wmma_kernel<KPAD, INF_><<<gemmBlocksN, T, 0, stream>>>(xp, Wl[0], nullptr, B1, (int)N);
        else
            gemm_wmma_kernel<HDIM, HDIM><<<gemmBlocksN, T, 0, stream>>>(B0, Wl[layer], nullptr, B1, (int)N);
        conv_init_kernel<<<nodeBlocks, T, 0, stream>>>(B1, dinv, bl[layer], B0, (int)N);
        conv_edge_kernel<<<edgeBlocks, T, 0, stream>>>(erow, ecol, ea, dinv, B1, B0, (int)E);
        if (layer < 3) {
            hipMemsetAsync(stats, 0, 2 * HDIM * sizeof(float), stream);
            stats_kernel<<<512, T, 0, stream>>>(B0, stats, N);
            bn_relu_kernel<<<(unsigned)cdiv(N * HDIM, T), T, 0, stream>>>(
                B0, B0, stats, bn_g + layer * HDIM, bn_b + layer * HDIM, N);
        }
    }

    // ---- global add pool ----
    hipMemsetAsync(gpool, 0, (size_t)G * HDIM * sizeof(float), stream);
    pool_kernel<<<nodeBlocks, T, 0, stream>>>(B0, batch, gpool, (int)N);

    // ---- MLP head: (g @ Wm0 + bm0) -> BN -> ReLU -> @ Wm1 + bm1 ----
    const unsigned gemmBlocksG = (unsigned)cdiv(cdiv((long long)G, 16), 8);
    gemm_wmma_kernel<HDIM, HDIM><<<gemmBlocksG, T, 0, stream>>>(gpool, Wm0, bm0, gw, G);
    hipMemsetAsync(stats, 0, 2 * HDIM * sizeof(float), stream);
    stats_kernel<<<8, T, 0, stream>>>(gw, stats, G);
    bn_relu_kernel<<<(unsigned)cdiv((long long)G * HDIM, T), T, 0, stream>>>(
        gw, gw, stats, bnm_g, bnm_b, G);
    head_kernel<<<(unsigned)cdiv((long long)G * CDIM, T), T, 0, stream>>>(
        gw, Wm1, bm1, (float*)d_out, G);
}